// SHNMN_36309653520744
// MI455X (gfx1250) — compile-verified
//
#include <hip/hip_runtime.h>
#include <math.h>

// ---------------- problem constants ----------------
#define BATCH   256
#define DIMC    64
#define E1      36928      // DIM + 64*64*9
#define E2      8256       // DIM + 2*64*64
#define QD      45184      // E1 + E2
#define CNNW    36864      // conv weight block in q_rep
#define CB_OFF  36864      // conv bias (64)
#define PW_OFF  36928      // proj weight (64x128)
#define PB_OFF  45120      // proj bias (64)
#define HW      256        // 16*16 spatial
#define FLAT    32768      // 512*8*8

// workspace offsets (floats)
#define OFF_CONST  0ll
#define OFF_QREP   64ll
#define OFF_X      (OFF_QREP + (long long)BATCH*QD)       // 11,567,168
#define OFF_H1     (OFF_X  + 4194304ll)
#define OFF_H2     (OFF_H1 + 4194304ll)
#define OFF_H3     (OFF_H2 + 4194304ll)
#define OFF_LHS    (OFF_H3 + 4194304ll)
#define OFF_RHS    (OFF_LHS + 4194304ll)
#define OFF_PROJ   (OFF_RHS + 4194304ll)
#define OFF_BIG    (OFF_PROJ + 4194304ll)                 // x1 (16.7M) then c (33.5M)
#define OFF_F      (OFF_BIG + 33554432ll)
#define OFF_FC1    (OFF_F + 8388608ll)

typedef float v2f __attribute__((ext_vector_type(2)));
typedef float v8f __attribute__((ext_vector_type(8)));

__device__ __forceinline__ v8f wmma4(v2f a, v2f b, v8f c) {
  // D = A(16x4,f32) x B(4x16,f32) + C(16x16,f32)
  return __builtin_amdgcn_wmma_f32_16x16x4_f32(false, a, false, b, (short)0, c,
                                               false, false);
}

// ---------------- tiny softmaxes ----------------
__global__ void shn_softmax_k(const float* __restrict__ alpha,
                              const float* __restrict__ tau0,
                              const float* __restrict__ tau1,
                              float* __restrict__ wsc) {
  if (threadIdx.x != 0 || blockIdx.x != 0) return;
  for (int i = 0; i < 3; ++i) {
    // alpha row i (8)
    float mx = -1e30f;
    for (int t = 0; t < 8; ++t) mx = fmaxf(mx, alpha[i * 8 + t]);
    float e[8], s = 0.f;
    for (int t = 0; t < 8; ++t) { e[t] = __expf(alpha[i * 8 + t] - mx); s += e[t]; }
    for (int t = 0; t < 8; ++t) wsc[i * 8 + t] = e[t] / s;
    int n = i + 2;
    // tau0 row i (first n of 4)
    mx = -1e30f;
    for (int j = 0; j < n; ++j) mx = fmaxf(mx, tau0[i * 4 + j]);
    s = 0.f; float e0[4];
    for (int j = 0; j < n; ++j) { e0[j] = __expf(tau0[i * 4 + j] - mx); s += e0[j]; }
    for (int j = 0; j < 4; ++j) wsc[24 + i * 4 + j] = (j < n) ? e0[j] / s : 0.f;
    // tau1 row i
    mx = -1e30f;
    for (int j = 0; j < n; ++j) mx = fmaxf(mx, tau1[i * 4 + j]);
    s = 0.f; float e1v[4];
    for (int j = 0; j < n; ++j) { e1v[j] = __expf(tau1[i * 4 + j] - mx); s += e1v[j]; }
    for (int j = 0; j < 4; ++j) wsc[36 + i * 4 + j] = (j < n) ? e1v[j] / s : 0.f;
  }
}

// ---------------- stem conv1: 3->64 on 64x64, relu, pool -> 32x32 ----------------
__global__ void shn_stem1_k(const float* __restrict__ img,
                            const float* __restrict__ W,
                            const float* __restrict__ bias,
                            float* __restrict__ X1) {
  int idx = blockIdx.x * 256 + threadIdx.x;
  if (idx >= BATCH * 64 * 32 * 32) return;
  int x = idx & 31, t = idx >> 5;
  int y = t & 31; t >>= 5;
  int o = t & 63; int b = t >> 6;
  float best = -1e30f;
  float bo = bias[o];
  for (int dy = 0; dy < 2; ++dy)
    for (int dx = 0; dx < 2; ++dx) {
      int yy = 2 * y + dy, xx = 2 * x + dx;
      float s = bo;
      #pragma unroll
      for (int c = 0; c < 3; ++c) {
        #pragma unroll
        for (int ky = 0; ky < 3; ++ky) {
          int iy = yy + ky - 1;
          if ((unsigned)iy >= 64u) continue;
          #pragma unroll
          for (int kx = 0; kx < 3; ++kx) {
            int ix = xx + kx - 1;
            if ((unsigned)ix >= 64u) continue;
            s += img[((b * 3 + c) * 64 + iy) * 64 + ix] *
                 W[((o * 3 + c) * 3 + ky) * 3 + kx];
          }
        }
      }
      best = fmaxf(best, s);
    }
  X1[idx] = fmaxf(best, 0.f);
}

// ---------------- stem conv2: 64->64 on 32x32, relu, pool -> 16x16 ----------------
__global__ void shn_stem2_k(const float* __restrict__ X1,
                            const float* __restrict__ W,
                            const float* __restrict__ bias,
                            float* __restrict__ X) {
  int idx = blockIdx.x * 256 + threadIdx.x;
  if (idx >= BATCH * 64 * 16 * 16) return;
  int x = idx & 15, t = idx >> 4;
  int y = t & 15; t >>= 4;
  int o = t & 63; int b = t >> 6;
  float bo = bias[o];
  float s00 = bo, s01 = bo, s10 = bo, s11 = bo;
  for (int c = 0; c < 64; ++c) {
    const float* base = X1 + ((long long)(b * 64 + c)) * 1024;
    float r[4][4];
    #pragma unroll
    for (int i = 0; i < 4; ++i) {
      int iy = 2 * y - 1 + i;
      #pragma unroll
      for (int j = 0; j < 4; ++j) {
        int ix = 2 * x - 1 + j;
        r[i][j] = ((unsigned)iy < 32u && (unsigned)ix < 32u) ? base[iy * 32 + ix] : 0.f;
      }
    }
    const float* w = W + (o * 64 + c) * 9;
    #pragma unroll
    for (int ky = 0; ky < 3; ++ky)
      #pragma unroll
      for (int kx = 0; kx < 3; ++kx) {
        float wv = w[ky * 3 + kx];
        s00 = fmaf(r[ky][kx],         wv, s00);
        s01 = fmaf(r[ky][kx + 1],     wv, s01);
        s10 = fmaf(r[ky + 1][kx],     wv, s10);
        s11 = fmaf(r[ky + 1][kx + 1], wv, s11);
      }
  }
  float best = fmaxf(fmaxf(s00, s01), fmaxf(s10, s11));
  X[idx] = fmaxf(best, 0.f);
}

// ---------------- q_rep: softmax-weighted gather of embeddings ----------------
__global__ void shn_qrep_k(const int* __restrict__ ques,
                           const float* __restrict__ e1,
                           const float* __restrict__ e2,
                           const float* __restrict__ wsc, int module,
                           float* __restrict__ Q) {
  int idx = blockIdx.x * 256 + threadIdx.x;
  if (idx >= BATCH * QD) return;
  int b = idx / QD;
  int d = idx - b * QD;
  const float* aw = wsc + module * 8;
  float s = 0.f;
  #pragma unroll
  for (int t = 0; t < 8; ++t) {
    int tok = ques[b * 8 + t];
    float v = (d < E1) ? e1[(long long)tok * E1 + d]
                       : e2[(long long)tok * E2 + (d - E1)];
    s = fmaf(aw[t], v, s);
  }
  Q[idx] = s;
}

// ---------------- state mix: lhs/rhs from tau-weighted states ----------------
__global__ void shn_mix_k(const float* __restrict__ wsc, int module,
                          const float* __restrict__ s0,
                          const float* __restrict__ s1,
                          const float* __restrict__ s2,
                          float* __restrict__ L, float* __restrict__ R) {
  int idx = blockIdx.x * 256 + threadIdx.x;
  if (idx >= BATCH * DIMC * HW) return;
  const float* t0 = wsc + 24 + module * 4;
  const float* t1 = wsc + 36 + module * 4;
  int ns = module + 1;
  float v = s0[idx];
  float l = t0[1] * v, r = t1[1] * v;
  if (ns >= 2) { float w = s1[idx]; l = fmaf(t0[2], w, l); r = fmaf(t1[2], w, r); }
  if (ns >= 3) { float w = s2[idx]; l = fmaf(t0[3], w, l); r = fmaf(t1[3], w, r); }
  L[idx] = l;
  R[idx] = r;
}

// ---------------- per-sample projection GEMM: 64x128 @ 128x256 (+pb) ----------------
__global__ void shn_proj_gemm(const float* __restrict__ Q,
                              const float* __restrict__ L,
                              const float* __restrict__ R,
                              float* __restrict__ P) {
  const int lane = threadIdx.x;
  const int b  = blockIdx.z;
  const int n0 = blockIdx.x * 16;   // hw tile
  const int m0 = blockIdx.y * 16;   // out-channel tile
  const int half = lane >> 4, l15 = lane & 15, k0 = half * 2;
  const float* pw = Q + (long long)b * QD + PW_OFF;
  const float* lb = L + b * (DIMC * HW);
  const float* rb = R + b * (DIMC * HW);
  v8f acc = {};
  for (int k = 0; k < 128; k += 4) {
    int kk = k + k0;
    v2f a; a.x = pw[(m0 + l15) * 128 + kk]; a.y = pw[(m0 + l15) * 128 + kk + 1];
    const float* src = (kk < 64) ? lb : rb;
    int kc = (kk < 64) ? kk : kk - 64;
    v2f bv; bv.x = src[kc * HW + n0 + l15]; bv.y = src[(kc + 1) * HW + n0 + l15];
    acc = wmma4(a, bv, acc);
  }
  const int col = n0 + l15;
  #pragma unroll
  for (int r = 0; r < 8; ++r) {
    int o = m0 + r + half * 8;
    P[b * (DIMC * HW) + o * HW + col] =
        acc[r] + Q[(long long)b * QD + PB_OFF + o];
  }
}

// ---------------- per-sample conv as implicit GEMM (K=576) ----------------
// proj tile (64 KB) staged memory->LDS with CDNA5 async-LDS loads (ASYNCcnt).
__global__ void shn_conv_ps(const float* __restrict__ Q,
                            const float* __restrict__ P,
                            float* __restrict__ Hout) {
  __shared__ float lds[DIMC * HW];      // 64 KB of the 320 KB WGP LDS
  const int b   = blockIdx.x;
  const int tid = threadIdx.x;
  const float* pbase = P + b * (DIMC * HW);

  // Async global->LDS: thread tid copies 16 B at LDS offset tid*16 + i*4096;
  // IOFFSET is added to both the global and LDS addresses (CDNA5 ISA),
  // so one LDS-offset VGPR + one 64-bit address pair covers all 16 transfers.
  {
    unsigned ldsa = (unsigned)(uintptr_t)(void*)lds + (unsigned)tid * 16u;
    unsigned long long ga = (unsigned long long)(uintptr_t)(pbase + tid * 4);
    #define SHN_ACP(OFF)                                                  \
      asm volatile("global_load_async_to_lds_b128 %0, %1, off offset:" #OFF \
                   :: "v"(ldsa), "v"(ga) : "memory")
    SHN_ACP(0);     SHN_ACP(4096);  SHN_ACP(8192);  SHN_ACP(12288);
    SHN_ACP(16384); SHN_ACP(20480); SHN_ACP(24576); SHN_ACP(28672);
    SHN_ACP(32768); SHN_ACP(36864); SHN_ACP(40960); SHN_ACP(45056);
    SHN_ACP(49152); SHN_ACP(53248); SHN_ACP(57344); SHN_ACP(61440);
    #undef SHN_ACP
    asm volatile("s_wait_asynccnt 0x0" ::: "memory");
  }
  __syncthreads();

  const int wave = tid >> 5;
  const int lane = tid & 31;
  const int half = lane >> 4, l15 = lane & 15, k0 = half * 2;
  const float* cw = Q + (long long)b * QD;   // [o][c*9+ky*3+kx]
  for (int t = wave; t < 64; t += 8) {       // 4 M-tiles x 16 N-tiles
    int mt = t >> 4, nt = t & 15;
    int m = mt * 16 + l15;
    int n = nt * 16 + l15;
    int y = n >> 4, x = n & 15;
    v8f acc = {};
    for (int k = 0; k < 576; k += 4) {
      int kk = k + k0;
      v2f a; a.x = cw[m * 576 + kk]; a.y = cw[m * 576 + kk + 1];
      v2f bv;
      {
        int c = kk / 9, tt = kk - c * 9;
        int ky = tt / 3, kx = tt - ky * 3;
        int iy = y + ky - 1, ix = x + kx - 1;
        bv.x = ((unsigned)iy < 16u && (unsigned)ix < 16u) ? lds[c * HW + iy * 16 + ix] : 0.f;
      }
      {
        int k2 = kk + 1;
        int c = k2 / 9, tt = k2 - c * 9;
        int ky = tt / 3, kx = tt - ky * 3;
        int iy = y + ky - 1, ix = x + kx - 1;
        bv.y = ((unsigned)iy < 16u && (unsigned)ix < 16u) ? lds[c * HW + iy * 16 + ix] : 0.f;
      }
      acc = wmma4(a, bv, acc);
    }
    #pragma unroll
    for (int r = 0; r < 8; ++r) {
      int o = mt * 16 + r + half * 8;
      float v = acc[r] + Q[(long long)b * QD + CB_OFF + o];
      Hout[b * (DIMC * HW) + o * HW + n] = fmaxf(v, 0.f);
    }
  }
}

// ---------------- head channel GEMM: 512x64 @ 64x256 per sample, +bias, relu ------
__global__ void shn_wproj_gemm(const float* __restrict__ W,
                               const float* __restrict__ bias,
                               const float* __restrict__ H,
                               float* __restrict__ C) {
  const int lane = threadIdx.x;
  const int b  = blockIdx.z;
  const int n0 = blockIdx.x * 16;   // hw tile (16)
  const int m0 = blockIdx.y * 16;   // o tile (32)
  const int half = lane >> 4, l15 = lane & 15, k0 = half * 2;
  const float* hb = H + b * (DIMC * HW);
  v8f acc = {};
  #pragma unroll
  for (int k = 0; k < 64; k += 4) {
    v2f a; a.x = W[(m0 + l15) * 64 + k + k0]; a.y = W[(m0 + l15) * 64 + k + k0 + 1];
    v2f bv; bv.x = hb[(k + k0) * HW + n0 + l15]; bv.y = hb[(k + k0 + 1) * HW + n0 + l15];
    acc = wmma4(a, bv, acc);
  }
  const int col = n0 + l15;
  #pragma unroll
  for (int r = 0; r < 8; ++r) {
    int o = m0 + r + half * 8;
    float v = acc[r] + bias[o];
    C[(long long)b * (512 * HW) + o * HW + col] = fmaxf(v, 0.f);
  }
}

// ---------------- 2x2 maxpool of head activations -> flat features ----------------
__global__ void shn_poolc_k(const float* __restrict__ C, float* __restrict__ F) {
  int idx = blockIdx.x * 256 + threadIdx.x;
  if (idx >= BATCH * 512 * 64) return;
  int x = idx & 7, t = idx >> 3;
  int y = t & 7; t >>= 3;
  int o = t & 511; int b = t >> 9;
  const float* base = C + ((long long)b * 512 + o) * HW;
  float v = fmaxf(fmaxf(base[(2 * y) * 16 + 2 * x],     base[(2 * y) * 16 + 2 * x + 1]),
                  fmaxf(base[(2 * y + 1) * 16 + 2 * x], base[(2 * y + 1) * 16 + 2 * x + 1]));
  F[idx] = v;
}

// ---------------- fc1: (256x32768) @ (32768x1024)^T + bias, relu ----------------
__global__ void shn_fc1_gemm(const float* __restrict__ F,
                             const float* __restrict__ W,
                             const float* __restrict__ bias,
                             float* __restrict__ out) {
  const int lane = threadIdx.x;
  const int n0 = blockIdx.x * 16;   // 64 tiles of 1024
  const int m0 = blockIdx.y * 16;   // 16 tiles of 256
  const int K = FLAT;
  const int half = lane >> 4, l15 = lane & 15, k0 = half * 2;
  const float* arow = F + (long long)(m0 + l15) * K + k0;
  const float* brow = W + (long long)(n0 + l15) * K + k0;
  v8f acc = {};
  for (int k = 0; k < K; k += 4) {
    if ((k & 511) == 0) {
      __builtin_prefetch(arow + k + 512, 0, 1);
      __builtin_prefetch(brow + k + 512, 0, 1);
    }
    v2f a; a.x = arow[k]; a.y = arow[k + 1];
    v2f b; b.x = brow[k]; b.y = brow[k + 1];
    acc = wmma4(a, b, acc);
  }
  const int col = n0 + l15;
  const float bb = bias[col];
  #pragma unroll
  for (int r = 0; r < 8; ++r) {
    int row = m0 + r + half * 8;
    float v = acc[r] + bb;
    out[row * 1024 + col] = fmaxf(v, 0.f);
  }
}

// ---------------- fc2: (256x1024) @ (1024x2)^T + bias ----------------
__global__ void shn_fc2_k(const float* __restrict__ F1,
                          const float* __restrict__ W,
                          const float* __restrict__ bias,
                          float* __restrict__ out) {
  int idx = blockIdx.x * 256 + threadIdx.x;
  if (idx >= BATCH * 2) return;
  int a = idx & 1, b = idx >> 1;
  float s = bias[a];
  for (int k = 0; k < 1024; ++k) s = fmaf(F1[b * 1024 + k], W[a * 1024 + k], s);
  out[b * 2 + a] = s;
}

extern "C" void kernel_launch(void* const* d_in, const int* in_sizes, int n_in,
                              void* d_out, int out_size, void* d_ws, size_t ws_size,
                              hipStream_t stream) {
  const float* image   = (const float*)d_in[0];
  const int*   question= (const int*)  d_in[1];
  const float* emb1    = (const float*)d_in[2];
  const float* emb2    = (const float*)d_in[3];
  const float* alpha   = (const float*)d_in[4];
  const float* tau0    = (const float*)d_in[5];
  const float* tau1    = (const float*)d_in[6];
  const float* w_s0    = (const float*)d_in[7];
  const float* b_s0    = (const float*)d_in[8];
  const float* w_s1    = (const float*)d_in[9];
  const float* b_s1    = (const float*)d_in[10];
  const float* w_proj  = (const float*)d_in[11];
  const float* b_proj  = (const float*)d_in[12];
  const float* w_fc1   = (const float*)d_in[13];
  const float* b_fc1   = (const float*)d_in[14];
  const float* w_fc2   = (const float*)d_in[15];
  const float* b_fc2   = (const float*)d_in[16];

  float* ws   = (float*)d_ws;
  float* wsc  = ws + OFF_CONST;
  float* qrep = ws + OFF_QREP;
  float* X    = ws + OFF_X;
  float* H[3] = { ws + OFF_H1, ws + OFF_H2, ws + OFF_H3 };
  float* LHS  = ws + OFF_LHS;
  float* RHS  = ws + OFF_RHS;
  float* PROJ = ws + OFF_PROJ;
  float* BIG  = ws + OFF_BIG;   // x1 first, later head activations c
  float* F    = ws + OFF_F;
  float* FC1  = ws + OFF_FC1;

  shn_softmax_k<<<1, 1, 0, stream>>>(alpha, tau0, tau1, wsc);
  shn_stem1_k<<<65536, 256, 0, stream>>>(image, w_s0, b_s0, BIG);
  shn_stem2_k<<<16384, 256, 0, stream>>>(BIG, w_s1, b_s1, X);

  for (int i = 0; i < 3; ++i) {
    shn_qrep_k<<<45184, 256, 0, stream>>>(question, emb1, emb2, wsc, i, qrep);
    shn_mix_k<<<16384, 256, 0, stream>>>(wsc, i, X, H[0], H[1], LHS, RHS);
    shn_proj_gemm<<<dim3(16, 4, BATCH), 32, 0, stream>>>(qrep, LHS, RHS, PROJ);
    shn_conv_ps<<<BATCH, 256, 0, stream>>>(qrep, PROJ, H[i]);
  }

  shn_wproj_gemm<<<dim3(16, 32, BATCH), 32, 0, stream>>>(w_proj, b_proj, H[2], BIG);
  shn_poolc_k<<<32768, 256, 0, stream>>>(BIG, F);
  shn_fc1_gemm<<<dim3(64, 16), 32, 0, stream>>>(F, w_fc1, b_fc1, FC1);
  shn_fc2_k<<<2, 256, 0, stream>>>(FC1, w_fc2, b_fc2, (float*)d_out);
}